// NonLocalBlack_30485677867283
// MI455X (gfx1250) — compile-verified
//
#include <hip/hip_runtime.h>
#include <hip/hip_bf16.h>
#include <math.h>

// Problem constants (BS=1)
#define C_IN  64
#define CH    32
#define NPOS  2304          // T*H*W = 4*24*24

typedef __attribute__((ext_vector_type(2))) float v2f;
typedef __attribute__((ext_vector_type(8))) float v8f;

// D = A(16x4 f32) * B(4x16 f32) + C(16x16 f32), wave32, EXEC all ones.
static __device__ __forceinline__ v8f wmma_f32_16x16x4(v2f a, v2f b, v8f c) {
    return __builtin_amdgcn_wmma_f32_16x16x4_f32(
        /*neg_a=*/false, a, /*neg_b=*/false, b,
        /*c_mod=*/(short)0, c, /*reuse_a=*/false, /*reuse_b=*/false);
}

// ---------------------------------------------------------------------------
// Kernel 1: v1/v2/gv (CH x N) = {theta,phi,g}_w (CH x C_IN) @ x (C_IN x N) + b
// One wave (32 threads) per 16x16 output tile; K=64 stepped by 4 (16 WMMAs x3).
// grid = (N/16, CH/16), block = 32
// ---------------------------------------------------------------------------
__global__ void nlb_proj_kernel(const float* __restrict__ x,
                                const float* __restrict__ tw, const float* __restrict__ tb,
                                const float* __restrict__ pw, const float* __restrict__ pb,
                                const float* __restrict__ gw, const float* __restrict__ gb,
                                float* __restrict__ v1, float* __restrict__ v2,
                                float* __restrict__ gvout) {
    const int lane  = threadIdx.x & 31;
    const int nbase = blockIdx.x * 16;
    const int mbase = blockIdx.y * 16;
    const int r     = lane & 15;          // M index for A, N index for B
    const int khalf = (lane >> 4) * 2;    // lanes 16-31 hold upper K pair

    v8f c1 = {}, c2 = {}, c3 = {};
    for (int kb = 0; kb < C_IN; kb += 4) {
        const int k0 = kb + khalf;
        v2f a1, a2, a3, b;
        a1.x = tw[(mbase + r) * C_IN + k0];
        a1.y = tw[(mbase + r) * C_IN + k0 + 1];
        a2.x = pw[(mbase + r) * C_IN + k0];
        a2.y = pw[(mbase + r) * C_IN + k0 + 1];
        a3.x = gw[(mbase + r) * C_IN + k0];
        a3.y = gw[(mbase + r) * C_IN + k0 + 1];
        b.x  = x[(k0    ) * NPOS + nbase + r];
        b.y  = x[(k0 + 1) * NPOS + nbase + r];
        c1 = wmma_f32_16x16x4(a1, b, c1);
        c2 = wmma_f32_16x16x4(a2, b, c2);
        c3 = wmma_f32_16x16x4(a3, b, c3);
    }
    // Store D: element (m = v + 8*(lane>>4), n = lane&15)
    const int n     = nbase + r;
    const int mhalf = (lane >> 4) * 8;
#pragma unroll
    for (int v = 0; v < 8; ++v) {
        const int m = mbase + mhalf + v;
        v1[m * NPOS + n]    = c1[v] + tb[m];
        v2[m * NPOS + n]    = c2[v] + pb[m];
        gvout[m * NPOS + n] = c3[v] + gb[m];
    }
}

// ---------------------------------------------------------------------------
// Kernel 2: per-channel min/max of v2 (for exact softmax row-max:
//   max_j a_i*b_j = a_i * (a_i>=0 ? max_j b_j : min_j b_j))
// grid = CH blocks of 256 threads
// ---------------------------------------------------------------------------
__global__ void nlb_minmax_kernel(const float* __restrict__ v2,
                                  float* __restrict__ minv, float* __restrict__ maxv) {
    __shared__ float smn[256];
    __shared__ float smx[256];
    const int c   = blockIdx.x;
    const int tid = threadIdx.x;
    float mn = INFINITY, mx = -INFINITY;
    for (int j = tid; j < NPOS; j += 256) {
        const float b = v2[c * NPOS + j];
        mn = fminf(mn, b);
        mx = fmaxf(mx, b);
    }
    smn[tid] = mn; smx[tid] = mx;
    __syncthreads();
    for (int s = 128; s > 0; s >>= 1) {
        if (tid < s) {
            smn[tid] = fminf(smn[tid], smn[tid + s]);
            smx[tid] = fmaxf(smx[tid], smx[tid + s]);
        }
        __syncthreads();
    }
    if (tid == 0) { minv[c] = smn[0]; maxv[c] = smx[0]; }
}

// ---------------------------------------------------------------------------
// Kernel 3: attention rows.  One wave per (c,i):
//   y[c,i] = sum_j exp(a*b_j - rm) * g_j / sum_j exp(a*b_j - rm)
// 256-thread blocks = 8 waves = 8 rows.  grid = CH*NPOS/8
// ---------------------------------------------------------------------------
__global__ void nlb_attn_kernel(const float* __restrict__ v1, const float* __restrict__ v2,
                                const float* __restrict__ gv,
                                const float* __restrict__ minv, const float* __restrict__ maxv,
                                float* __restrict__ y) {
    const int wave = threadIdx.x >> 5;
    const int lane = threadIdx.x & 31;
    const int row  = blockIdx.x * 8 + wave;   // 0 .. CH*NPOS-1
    const int c    = row / NPOS;
    const int i    = row - c * NPOS;

    const float a  = v1[c * NPOS + i];
    const float rm = (a >= 0.f) ? a * maxv[c] : a * minv[c];

    const float* __restrict__ B = v2 + c * NPOS;
    const float* __restrict__ G = gv + c * NPOS;

    float num = 0.f, den = 0.f;
    for (int j = lane; j < NPOS; j += 32) {     // 72 iters, fully coalesced
        const float e = __expf(fmaf(a, B[j], -rm));
        num = fmaf(e, G[j], num);
        den += e;
    }
    // wave32 butterfly reduction
#pragma unroll
    for (int off = 16; off > 0; off >>= 1) {
        num += __shfl_xor(num, off, 32);
        den += __shfl_xor(den, off, 32);
    }
    if (lane == 0) y[c * NPOS + i] = num / den;
}

// ---------------------------------------------------------------------------
// Kernel 4: out (C_IN x N) = x + k_w (C_IN x CH) @ y (CH x N) + k_b
// One wave per 16x16 tile, K=32 stepped by 4.  grid = (N/16, C_IN/16)
// ---------------------------------------------------------------------------
__global__ void nlb_outproj_kernel(const float* __restrict__ kw, const float* __restrict__ kb,
                                   const float* __restrict__ y, const float* __restrict__ x,
                                   float* __restrict__ out) {
    const int lane  = threadIdx.x & 31;
    const int nbase = blockIdx.x * 16;
    const int mbase = blockIdx.y * 16;
    const int r     = lane & 15;
    const int khalf = (lane >> 4) * 2;

    v8f acc = {};
    for (int kbb = 0; kbb < CH; kbb += 4) {
        const int k0 = kbb + khalf;
        v2f a, b;
        a.x = kw[(mbase + r) * CH + k0];
        a.y = kw[(mbase + r) * CH + k0 + 1];
        b.x = y[(k0    ) * NPOS + nbase + r];
        b.y = y[(k0 + 1) * NPOS + nbase + r];
        acc = wmma_f32_16x16x4(a, b, acc);
    }
    const int n     = nbase + r;
    const int mhalf = (lane >> 4) * 8;
#pragma unroll
    for (int v = 0; v < 8; ++v) {
        const int m = mbase + mhalf + v;
        out[m * NPOS + n] = x[m * NPOS + n] + acc[v] + kb[m];
    }
}

// ---------------------------------------------------------------------------
// Launcher.  Workspace layout (floats):
//   v1: [0, 73728)  v2: [73728, 147456)  gv: [147456, 221184)
//   y:  [221184, 294912)  minv: [294912, 294944)  maxv: [294944, 294976)
// Total ~1.18 MB.
// ---------------------------------------------------------------------------
extern "C" void kernel_launch(void* const* d_in, const int* in_sizes, int n_in,
                              void* d_out, int out_size, void* d_ws, size_t ws_size,
                              hipStream_t stream) {
    const float* x  = (const float*)d_in[0];
    const float* tw = (const float*)d_in[1];
    const float* tb = (const float*)d_in[2];
    const float* pw = (const float*)d_in[3];
    const float* pb = (const float*)d_in[4];
    const float* gw = (const float*)d_in[5];
    const float* gb = (const float*)d_in[6];
    const float* kw = (const float*)d_in[7];
    const float* kb = (const float*)d_in[8];
    float* out = (float*)d_out;

    float* ws   = (float*)d_ws;
    float* v1   = ws;
    float* v2   = ws + (size_t)CH * NPOS;
    float* gv   = ws + (size_t)2 * CH * NPOS;
    float* y    = ws + (size_t)3 * CH * NPOS;
    float* minv = ws + (size_t)4 * CH * NPOS;
    float* maxv = minv + CH;

    // 1) projections: 144 n-tiles x 2 m-tiles, one wave per tile
    nlb_proj_kernel<<<dim3(NPOS / 16, CH / 16), 32, 0, stream>>>(
        x, tw, tb, pw, pb, gw, gb, v1, v2, gv);

    // 2) per-channel min/max of v2
    nlb_minmax_kernel<<<CH, 256, 0, stream>>>(v2, minv, maxv);

    // 3) attention: one wave per (c,i) row
    nlb_attn_kernel<<<(CH * NPOS) / 8, 256, 0, stream>>>(v1, v2, gv, minv, maxv, y);

    // 4) output projection + residual
    nlb_outproj_kernel<<<dim3(NPOS / 16, C_IN / 16), 32, 0, stream>>>(
        kw, kb, y, x, out);
}